// Gradient_block_50276887166993
// MI455X (gfx1250) — compile-verified
//
#include <hip/hip_runtime.h>
#include <hip/hip_bf16.h>

typedef __attribute__((ext_vector_type(2))) float v2f;
typedef __attribute__((ext_vector_type(8))) float v8f;

#define TILES_PER_WAVE 4

// ---- constants of the fused Jacobian (see analysis) ----
static constexpr float kC0  = 0.19947114020071634f;   // sqrt(1/(8*pi))      : f0 -> diagonal
static constexpr float kC1  = 0.14104739588693907f;   // 1/(4*sqrt(pi))      : f1 -> antisymmetric
static constexpr float kS10 = 0.10925484305920792f;   // sqrt(3/(8pi))/sqrt(10) : f2 off-diag
static constexpr float kS30 = 0.06307831305050401f;   // sqrt(3/(8pi))/sqrt(30) : f2 diag

// W is the constant 32x9 matrix with out[b,p] = sum_k nf[b,k] * W[k][p].
// Only K-chunks {0,1,2,4,5,6} (of 4) are nonzero -> 6 WMMA steps.
// Stored pre-interleaved for the B-matrix VGPR layout of V_WMMA_F32_16X16X4_F32:
//   v[t][h][n][e] = W[4*c(t) + 2*h + e][n],  c(t) in {0,1,2,4,5,6}
struct alignas(16) WTab { float v[6][2][16][2]; };

constexpr void setw(WTab& w, int k, int p, float val) {
    int c  = k >> 2;
    int t  = (c < 4) ? c : c - 1;      // chunk c in {0,1,2,4,5,6} -> slot t in 0..5
    int kk = k & 3;
    w.v[t][kk >> 1][p][kk & 1] = val;
}

constexpr WTab makeW() {
    WTab w{};
    // f0 paths (k = 0,1): c0 on diagonal p = 0,4,8
    for (int k = 0; k <= 1; ++k) {
        setw(w, k, 0, kC0); setw(w, k, 4, kC0); setw(w, k, 8, kC0);
    }
    // f1 paths (k = 4..9 = two copies of f1[0..2]): antisymmetric eps term
    for (int d = 0; d <= 3; d += 3) {
        setw(w, 4 + d, 5, -kC1); setw(w, 4 + d, 7,  kC1);   // f1[0]
        setw(w, 5 + d, 2,  kC1); setw(w, 5 + d, 6, -kC1);   // f1[1]
        setw(w, 6 + d, 1, -kC1); setw(w, 6 + d, 3,  kC1);   // f1[2]
    }
    // f2 paths (k = 16..25 = two copies of f2[0..4]): symmetric w3j(2,1,1) term
    for (int d = 0; d <= 5; d += 5) {
        setw(w, 16 + d, 2,  kS10); setw(w, 16 + d, 6,  kS10);                      // f2[0]
        setw(w, 17 + d, 1,  kS10); setw(w, 17 + d, 3,  kS10);                      // f2[1]
        setw(w, 18 + d, 0, -kS30); setw(w, 18 + d, 4, 2.0f * kS30);
        setw(w, 18 + d, 8, -kS30);                                                 // f2[2]
        setw(w, 19 + d, 5,  kS10); setw(w, 19 + d, 7,  kS10);                      // f2[3]
        setw(w, 20 + d, 0, -kS10); setw(w, 20 + d, 8,  kS10);                      // f2[4]
    }
    return w;
}

__device__ constexpr WTab g_W = makeW();

// One wave computes a 16-row tile: D(16x16) = A(16x32) * W(32x16), K split in 6
// nonzero chunks of 4, on the f32 WMMA pipe. Memory-bound: one 128B line per row in,
// 36B per row out.
__global__ __launch_bounds__(256) void sh_tp_grad_wmma(const float* __restrict__ nf,
                                                       float* __restrict__ out,
                                                       int numTiles) {
    const int lane = threadIdx.x & 31;
    const int n    = lane & 15;    // A: row-in-tile / D: column index
    const int h    = lane >> 4;    // half-wave selector (K pair / M+8)
    const int wave = blockIdx.x * (blockDim.x >> 5) + (threadIdx.x >> 5);

    // B fragments (constant W), one B64 per chunk; table is L0/L2 resident.
    v2f bmat[6];
    {
        const float* wbase = &g_W.v[0][0][0][0];
#pragma unroll
        for (int t = 0; t < 6; ++t)
            bmat[t] = *(const v2f*)(wbase + ((size_t)((t * 2 + h) * 16 + n)) * 2);
    }
    const int koff[6] = {0, 4, 8, 16, 20, 24};

#pragma unroll 1
    for (int s = 0; s < TILES_PER_WAVE; ++s) {
        const int tile = wave * TILES_PER_WAVE + s;
        if (tile >= numTiles) return;                  // wave-uniform guard
        const long baseRow = (long)tile * 16;

        // A fragment source: lane(h,n) reads floats (koff+2h, koff+2h+1) of row baseRow+n
        const float* ap = nf + (size_t)(baseRow + n) * 64 + 2 * h;

        v8f acc = {};
#pragma unroll
        for (int t = 0; t < 6; ++t) {
            v2f a = *(const v2f*)(ap + koff[t]);
            acc = __builtin_amdgcn_wmma_f32_16x16x4_f32(
                /*neg_a=*/false, a, /*neg_b=*/false, bmat[t],
                /*c_mod=*/(short)0, acc, /*reuse_a=*/false, /*reuse_b=*/false);
        }

        // D layout: VGPR v, lane(16h+n) holds out[baseRow + v + 8h, n]; columns 9..15 are 0.
        if (n < 9) {
            float* op = out + (size_t)(baseRow + 8 * h) * 9 + n;
#pragma unroll
            for (int v = 0; v < 8; ++v)
                op[(size_t)v * 9] = acc[v];
        }
    }
}

// Scalar fallback for B % 16 leftover rows (not hit for B = 1M, kept for robustness).
__global__ void sh_tp_grad_tail(const float* __restrict__ nf, float* __restrict__ out,
                                long startRow, int nRows) {
    int i = blockIdx.x * blockDim.x + threadIdx.x;
    if (i >= nRows) return;
    const float* x = nf + (size_t)(startRow + i) * 64;
    float f0  = x[0] + x[1];
    float f10 = x[4] + x[7], f11 = x[5] + x[8], f12 = x[6] + x[9];
    float f20 = x[16] + x[21], f21 = x[17] + x[22], f22 = x[18] + x[23];
    float f23 = x[19] + x[24], f24 = x[20] + x[25];
    float* o = out + (size_t)(startRow + i) * 9;
    o[0] =  kC0 * f0 - kS30 * f22 - kS10 * f24;
    o[1] = -kC1 * f12 + kS10 * f21;
    o[2] =  kC1 * f11 + kS10 * f20;
    o[3] =  kC1 * f12 + kS10 * f21;
    o[4] =  kC0 * f0 + 2.0f * kS30 * f22;
    o[5] = -kC1 * f10 + kS10 * f23;
    o[6] = -kC1 * f11 + kS10 * f20;
    o[7] =  kC1 * f10 + kS10 * f23;
    o[8] =  kC0 * f0 - kS30 * f22 + kS10 * f24;
}

extern "C" void kernel_launch(void* const* d_in, const int* in_sizes, int n_in,
                              void* d_out, int out_size, void* d_ws, size_t ws_size,
                              hipStream_t stream) {
    (void)n_in; (void)out_size; (void)d_ws; (void)ws_size;
    const float* nf = (const float*)d_in[0];
    float* out = (float*)d_out;

    const long B = (long)in_sizes[0] / 64;
    const int numTiles = (int)(B / 16);
    const int rem = (int)(B % 16);

    if (numTiles > 0) {
        const int waves  = (numTiles + TILES_PER_WAVE - 1) / TILES_PER_WAVE;
        const int blocks = (waves + 7) / 8;              // 8 waves (256 threads) per block
        sh_tp_grad_wmma<<<blocks, 256, 0, stream>>>(nf, out, numTiles);
    }
    if (rem > 0) {
        sh_tp_grad_tail<<<1, 32, 0, stream>>>(nf, out, (long)numTiles * 16, rem);
    }
}